// LightGCN2_68453188764174
// MI455X (gfx1250) — compile-verified
//
#include <hip/hip_runtime.h>
#include <hip/hip_bf16.h>

#define USERS  100000
#define ITEMS  50000
#define NNODES (USERS + ITEMS)
#define FEAT   256
#define HID    128

typedef float v2f __attribute__((ext_vector_type(2)));
typedef float v8f __attribute__((ext_vector_type(8)));

// ---------------------------------------------------------------------------
// utility kernels
// ---------------------------------------------------------------------------
__global__ void zero4_kernel(float* __restrict__ p, int n4) {
  int i = blockIdx.x * blockDim.x + threadIdx.x;
  if (i < n4) ((float4*)p)[i] = make_float4(0.f, 0.f, 0.f, 0.f);
}

__global__ void copy4_kernel(float* __restrict__ out, const float* __restrict__ in, int n4) {
  int i = blockIdx.x * blockDim.x + threadIdx.x;
  if (i < n4) ((float4*)out)[i] = ((const float4*)in)[i];
}

__global__ void axpy4_kernel(float* __restrict__ out, const float* __restrict__ in,
                             float scale, int n4) {
  int i = blockIdx.x * blockDim.x + threadIdx.x;
  if (i < n4) {
    float4 a = ((float4*)out)[i];
    float4 b = ((const float4*)in)[i];
    a.x += b.x * scale; a.y += b.y * scale; a.z += b.z * scale; a.w += b.w * scale;
    ((float4*)out)[i] = a;
  }
}

// ---------------------------------------------------------------------------
// degree / normalization
// ---------------------------------------------------------------------------
__global__ void degree_kernel(const int* __restrict__ src, const int* __restrict__ dst,
                              float* __restrict__ degO, float* __restrict__ degI, int E) {
  int i = blockIdx.x * blockDim.x + threadIdx.x;
  if (i < E) {
    atomicAdd(&degO[src[i]], 1.0f);
    atomicAdd(&degI[dst[i]], 1.0f);
  }
}

__global__ void invsqrt_kernel(float* __restrict__ d, int n) {
  int i = blockIdx.x * blockDim.x + threadIdx.x;
  if (i < n) d[i] = rsqrtf(fmaxf(d[i], 1.0f));
}

// ---------------------------------------------------------------------------
// fp32 WMMA GEMM: C[rows x HID] = act(A[rows x K] @ B[K x HID] + bias)
// one wave -> one 16x16 output tile; block = 8 waves = 8 column tiles (HID=128)
// grid.x = rows/16
// ---------------------------------------------------------------------------
template <bool RELU>
__global__ __launch_bounds__(256) void mlp_gemm_kernel(
    const float* __restrict__ A, const float* __restrict__ B,
    const float* __restrict__ bias, float* __restrict__ C, int K) {
  const int lane = threadIdx.x & 31;
  const int cw   = threadIdx.x >> 5;   // column tile 0..7
  const int half = lane >> 4;          // which half-wave
  const int l    = lane & 15;
  const int row  = blockIdx.x * 16 + l;    // A row this lane feeds (M = l)
  const int col  = cw * 16 + l;            // output column this lane holds (N = l)
  const float* __restrict__ arow = A + (size_t)row * K;

  v8f acc = {};
  for (int k0 = 0; k0 < K; k0 += 4) {
    const int ka = k0 + 2 * half;  // lanes 0-15: K=k0,k0+1 ; lanes 16-31: K=k0+2,k0+3
    v2f a = *(const v2f*)(arow + ka);                 // A[M=row][ka], A[M=row][ka+1]
    v2f b;
    b.x = B[(size_t)ka * HID + col];                  // B[K=ka][N=col]
    b.y = B[(size_t)(ka + 1) * HID + col];            // B[K=ka+1][N=col]
    acc = __builtin_amdgcn_wmma_f32_16x16x4_f32(
        /*neg_a=*/false, a, /*neg_b=*/false, b,
        /*c_mod=*/(short)0, acc, /*reuse_a=*/false, /*reuse_b=*/false);
  }

  const float bv = bias[col];
#pragma unroll
  for (int r = 0; r < 8; ++r) {
    // C/D layout: VGPR r -> M = r (lanes 0-15) / M = r+8 (lanes 16-31), N = l
    const int m = blockIdx.x * 16 + r + 8 * half;
    float v = acc[r] + bv;
    if (RELU) v = fmaxf(v, 0.0f);
    C[(size_t)m * HID + col] = v;
  }
}

// ---------------------------------------------------------------------------
// edge aggregation: one wave per edge, 32 lanes x float4 = 128 feats
// agg[dst] += node_f[src] * (inv_out[src] * inv_in[dst])
// ---------------------------------------------------------------------------
__global__ __launch_bounds__(256) void edge_agg_kernel(
    const int* __restrict__ src, const int* __restrict__ dst,
    const float* __restrict__ invO, const float* __restrict__ invI,
    const float* __restrict__ nodef, float* __restrict__ agg, int E) {
  const int e    = (blockIdx.x * blockDim.x + threadIdx.x) >> 5;
  const int lane = threadIdx.x & 31;
  if (e >= E) return;
  const int s = src[e];
  const int d = dst[e];
  const float w = invO[s] * invI[d];
  float4 v = *(const float4*)(nodef + (size_t)s * HID + lane * 4);
  float* ap = agg + (size_t)d * HID + lane * 4;
  atomicAdd(ap + 0, v.x * w);
  atomicAdd(ap + 1, v.y * w);
  atomicAdd(ap + 2, v.z * w);
  atomicAdd(ap + 3, v.w * w);
}

// ---------------------------------------------------------------------------
// launcher
// ---------------------------------------------------------------------------
extern "C" void kernel_launch(void* const* d_in, const int* in_sizes, int n_in,
                              void* d_out, int out_size, void* d_ws, size_t ws_size,
                              hipStream_t stream) {
  const float* item_feat = (const float*)d_in[0];
  const float* user_emb  = (const float*)d_in[1];
  const float* W1        = (const float*)d_in[2];
  const float* b1        = (const float*)d_in[3];
  const float* W2        = (const float*)d_in[4];
  const float* b2        = (const float*)d_in[5];
  const int*   src       = (const int*)d_in[6];
  const int*   dst       = (const int*)d_in[7];
  const int    E         = in_sizes[6];

  float* out      = (float*)d_out;
  float* res_user = out;                          // [USERS x HID]
  float* res_item = out + (size_t)USERS * HID;    // [ITEMS x HID]

  // workspace layout
  float* invO = (float*)d_ws;                     // N
  float* invI = invO + NNODES;                    // N
  float* bufA = invI + NNODES;                    // N*HID
  float* bufB = bufA + (size_t)NNODES * HID;      // N*HID
  float* Hbuf = bufB;                             // ITEMS*HID, aliases bufB (dead before bufB use)

  const int nf4 = NNODES * HID / 4;               // float4 count for a node-feature buffer
  const int TB  = 256;

  // --- degrees -> inv-sqrt norms (invO/invI contiguous: zero & rsqrt together)
  zero4_kernel<<<(2 * NNODES / 4 + TB - 1) / TB, TB, 0, stream>>>(invO, 2 * NNODES / 4);
  degree_kernel<<<(E + TB - 1) / TB, TB, 0, stream>>>(src, dst, invO, invI, E);
  invsqrt_kernel<<<(2 * NNODES + TB - 1) / TB, TB, 0, stream>>>(invO, 2 * NNODES);

  // --- item MLP: Hbuf = relu(item_feat @ W1 + b1); res_item = Hbuf @ W2 + b2
  mlp_gemm_kernel<true ><<<ITEMS / 16, TB, 0, stream>>>(item_feat, W1, b1, Hbuf, FEAT);
  mlp_gemm_kernel<false><<<ITEMS / 16, TB, 0, stream>>>(Hbuf, W2, b2, res_item, HID);

  // --- res_user = user_emb
  copy4_kernel<<<(USERS * HID / 4 + TB - 1) / TB, TB, 0, stream>>>(
      res_user, user_emb, USERS * HID / 4);

  const int eblocks = (E + 7) / 8;  // 8 waves per block, 1 edge per wave
  const int zblocks = (nf4 + TB - 1) / TB;

  // --- layer 0: node_f = [res_user; res_item] == d_out
  zero4_kernel<<<zblocks, TB, 0, stream>>>(bufA, nf4);
  edge_agg_kernel<<<eblocks, TB, 0, stream>>>(src, dst, invO, invI, out, bufA, E);
  axpy4_kernel<<<zblocks, TB, 0, stream>>>(out, bufA, 1.0f / 2.0f, nf4);

  // --- layer 1: node_f = emb(layer0) = bufA
  zero4_kernel<<<zblocks, TB, 0, stream>>>(bufB, nf4);
  edge_agg_kernel<<<eblocks, TB, 0, stream>>>(src, dst, invO, invI, bufA, bufB, E);
  axpy4_kernel<<<zblocks, TB, 0, stream>>>(out, bufB, 1.0f / 3.0f, nf4);

  // --- layer 2: node_f = emb(layer1) = bufB
  zero4_kernel<<<zblocks, TB, 0, stream>>>(bufA, nf4);
  edge_agg_kernel<<<eblocks, TB, 0, stream>>>(src, dst, invO, invI, bufB, bufA, E);
  axpy4_kernel<<<zblocks, TB, 0, stream>>>(out, bufA, 1.0f / 4.0f, nf4);
}